// SDF_39264591020352
// MI455X (gfx1250) — compile-verified
//
#include <hip/hip_runtime.h>
#include <math.h>
#include <stdint.h>

#define TILE 256          // LDS tile capacity (faces); runtime face-block size <= TILE
#define BLK  256          // threads per block = 8 wave32
#define EPSF 1e-4f

// ---------- NaN-first argmin semantics (numpy/JAX): first NaN wins, ties keep earlier ----------
__device__ __forceinline__ bool better(float n, float c) {
    return (c == c) && (!(n == n) || (n < c));
}

// NaN-preserving clamp to [0,1] (jnp.clip propagates NaN; fminf/fmaxf would not)
__device__ __forceinline__ float clamp01_nan(float s) {
    s = (s < 0.0f) ? 0.0f : s;
    s = (s > 1.0f) ? 1.0f : s;
    return s;
}

// ---------- CDNA5 async global->LDS copy (gfx1250) ----------
__device__ __forceinline__ void async_b32(unsigned lds_byte_addr, const float* g) {
    asm volatile("global_load_async_to_lds_b32 %0, %1, off"
                 :: "v"(lds_byte_addr), "v"((unsigned long long)(uintptr_t)g)
                 : "memory");
}
__device__ __forceinline__ void wait_async0() {
    asm volatile("s_wait_asynccnt 0" ::: "memory");
}

// Gather one tile of faces (a,b,c vertices) into LDS via async-to-LDS, then
// precompute 1/|e|^2 for the three edges. Record stride = 16 floats (64B, b128-friendly).
__device__ __forceinline__ void load_face_tile(const float* __restrict__ v,
                                               const int* __restrict__ f,
                                               int faceBase, int cnt,
                                               float* sface) {
    const int t = threadIdx.x;
    if (t < cnt) {
        const int j  = faceBase + t;
        const int i0 = f[3 * j + 0];
        const int i1 = f[3 * j + 1];
        const int i2 = f[3 * j + 2];
        const unsigned lb = (unsigned)(uintptr_t)(&sface[t * 16]);
        const float* va = v + 3 * (long)i0;
        const float* vb = v + 3 * (long)i1;
        const float* vc = v + 3 * (long)i2;
        async_b32(lb + 0,  va + 0); async_b32(lb + 4,  va + 1); async_b32(lb + 8,  va + 2);
        async_b32(lb + 12, vb + 0); async_b32(lb + 16, vb + 1); async_b32(lb + 20, vb + 2);
        async_b32(lb + 24, vc + 0); async_b32(lb + 28, vc + 1); async_b32(lb + 32, vc + 2);
    }
    wait_async0();                    // this wave's async writes now visible to itself
    if (t < cnt) {
        const float ax = sface[t*16+0], ay = sface[t*16+1], az = sface[t*16+2];
        const float bx = sface[t*16+3], by = sface[t*16+4], bz = sface[t*16+5];
        const float cx = sface[t*16+6], cy = sface[t*16+7], cz = sface[t*16+8];
        const float e1x = ax-bx, e1y = ay-by, e1z = az-bz;
        const float e2x = cx-bx, e2y = cy-by, e2z = cz-bz;
        const float e3x = cx-ax, e3y = cy-ay, e3z = cz-az;
        sface[t*16+9]  = 1.0f / (e1x*e1x + e1y*e1y + e1z*e1z);  // inf for degenerate -> 0*inf=NaN like 0/0
        sface[t*16+10] = 1.0f / (e2x*e2x + e2y*e2y + e2z*e2z);
        sface[t*16+11] = 1.0f / (e3x*e3x + e3y*e3y + e3z*e3z);
    }
    __syncthreads();                  // cross-wave visibility of tile
}

// Inner loop: one thread owns one point, scans faces of the tile (LDS broadcast reads).
__device__ __forceinline__ void process_tile(const float* sface, int cnt,
                                             float px, float py, float pz,
                                             float dvx, float dvy, float dvz,
                                             float& best, float& bcx, float& bcy, float& bcz,
                                             int& hits) {
    for (int j = 0; j < cnt; ++j) {
        const float* fr = &sface[j * 16];
        const float ax = fr[0], ay = fr[1], az = fr[2];
        const float bx = fr[3], by = fr[4], bz = fr[5];
        const float cx = fr[6], cy = fr[7], cz = fr[8];
        const float i1 = fr[9], i2 = fr[10], i3 = fr[11];

        const float apx = ax-px, apy = ay-py, apz = az-pz;
        const float cpx = cx-px, cpy = cy-py, cpz = cz-pz;

        // seg(e1=a-b, ref=a, q0=b, q1=a)
        const float e1x = ax-bx, e1y = ay-by, e1z = az-bz;
        const float s1 = clamp01_nan((apx*e1x + apy*e1y + apz*e1z) * i1);
        const float w1 = 1.0f - s1;
        const float c1x = s1*bx + w1*ax, c1y = s1*by + w1*ay, c1z = s1*bz + w1*az;
        float gx = c1x-px, gy = c1y-py, gz = c1z-pz;
        const float d1 = gx*gx + gy*gy + gz*gz;

        // seg(e2=c-b, ref=c, q0=b, q1=c)
        const float e2x = cx-bx, e2y = cy-by, e2z = cz-bz;
        const float s2 = clamp01_nan((cpx*e2x + cpy*e2y + cpz*e2z) * i2);
        const float w2 = 1.0f - s2;
        const float c2x = s2*bx + w2*cx, c2y = s2*by + w2*cy, c2z = s2*bz + w2*cz;
        gx = c2x-px; gy = c2y-py; gz = c2z-pz;
        const float d2 = gx*gx + gy*gy + gz*gz;

        // seg(e3=c-a, ref=c, q0=a, q1=c)
        const float e3x = cx-ax, e3y = cy-ay, e3z = cz-az;
        const float s3 = clamp01_nan((cpx*e3x + cpy*e3y + cpz*e3z) * i3);
        const float w3 = 1.0f - s3;
        const float c3x = s3*ax + w3*cx, c3y = s3*ay + w3*cy, c3z = s3*az + w3*cz;
        gx = c3x-px; gy = c3y-py; gz = c3z-pz;
        const float d3 = gx*gx + gy*gy + gz*gz;

        float fsq = d1, fx = c1x, fy = c1y, fz = c1z;
        if (better(d2, fsq)) { fsq = d2; fx = c2x; fy = c2y; fz = c2z; }
        if (better(d3, fsq)) { fsq = d3; fx = c3x; fy = c3y; fz = c3z; }
        if (better(fsq, best)) { best = fsq; bcx = fx; bcy = fy; bcz = fz; }

        // Moller-Trumbore along dirv = vmean - p
        const float m1x = bx-ax, m1y = by-ay, m1z = bz-az;
        const float m2x = cx-ax, m2y = cy-ay, m2z = cz-az;
        const float pvx = dvy*m2z - dvz*m2y;
        const float pvy = dvz*m2x - dvx*m2z;
        const float pvz = dvx*m2y - dvy*m2x;
        const float det = m1x*pvx + m1y*pvy + m1z*pvz;
        if (fabsf(det) > EPSF) {
            const float tvx = px-ax, tvy = py-ay, tvz = pz-az;
            const float u   = (tvx*pvx + tvy*pvy + tvz*pvz) / det;
            const float qvx = tvy*m1z - tvz*m1y;
            const float qvy = tvz*m1x - tvx*m1z;
            const float qvz = tvx*m1y - tvy*m1x;
            const float dd  = (dvx*qvx + dvy*qvy + dvz*qvz) / det;
            const float tt  = (m2x*qvx + m2y*qvy + m2z*qvz) / det;
            if (u >= 0.0f && u <= 1.0f && dd >= 0.0f && (u + dd) <= 1.0f && tt > EPSF) hits++;
        }
    }
}

// ---------- kernel 0: vertex mean -> ws[0..2] ----------
__global__ void sdf_vmean(const float* __restrict__ v, float* __restrict__ ws, int V) {
    __shared__ float sx[BLK], sy[BLK], sz[BLK];
    const int t = threadIdx.x;
    float ax = 0.f, ay = 0.f, az = 0.f;
    for (int j = t; j < V; j += BLK) { ax += v[3*j]; ay += v[3*j+1]; az += v[3*j+2]; }
    sx[t] = ax; sy[t] = ay; sz[t] = az;
    __syncthreads();
    for (int off = BLK / 2; off > 0; off >>= 1) {
        if (t < off) { sx[t] += sx[t+off]; sy[t] += sy[t+off]; sz[t] += sz[t+off]; }
        __syncthreads();
    }
    if (t == 0) {
        ws[0] = sx[0] / (float)V;
        ws[1] = sy[0] / (float)V;
        ws[2] = sz[0] / (float)V;
    }
}

// ---------- pass 1: grid (pointBlocks x faceBlocks), partials to workspace ----------
__global__ void sdf_pass1(const float* __restrict__ p, const float* __restrict__ v,
                          const int* __restrict__ f, float* __restrict__ ws,
                          int N, int F, int fbs, int nfb) {
    __shared__ __align__(16) float sface[TILE * 16];
    const int t  = threadIdx.x;
    const int fb = blockIdx.y;
    const int faceBase = fb * fbs;
    int cnt = F - faceBase; if (cnt > fbs) cnt = fbs;

    load_face_tile(v, f, faceBase, cnt, sface);

    const int i = blockIdx.x * BLK + t;
    const bool valid = (i < N);
    const float px = valid ? p[3*i]   : 0.f;
    const float py = valid ? p[3*i+1] : 0.f;
    const float pz = valid ? p[3*i+2] : 0.f;
    const float dvx = ws[0] - px, dvy = ws[1] - py, dvz = ws[2] - pz;

    float best = __builtin_inff(), bcx = 0.f, bcy = 0.f, bcz = 0.f;
    int hits = 0;
    process_tile(sface, cnt, px, py, pz, dvx, dvy, dvz, best, bcx, bcy, bcz, hits);

    if (valid) {
        float4* rec = (float4*)(ws + 16 + ((size_t)i * nfb + fb) * 8);   // 32B-aligned
        rec[0] = make_float4(best, bcx, bcy, bcz);
        rec[1] = make_float4(__int_as_float(hits), 0.f, 0.f, 0.f);
    }
}

// ---------- pass 2: reduce face-block partials in ascending order ----------
__global__ void sdf_pass2(const float* __restrict__ ws, float* __restrict__ out,
                          int N, int nfb) {
    const int i = blockIdx.x * blockDim.x + threadIdx.x;
    if (i >= N) return;
    float best = __builtin_inff(), bx = 0.f, by = 0.f, bz = 0.f;
    int hits = 0;
    for (int fb = 0; fb < nfb; ++fb) {
        const float4* rec = (const float4*)(ws + 16 + ((size_t)i * nfb + fb) * 8);
        const float4 lo = rec[0];
        const float4 hi = rec[1];
        if (better(lo.x, best)) { best = lo.x; bx = lo.y; by = lo.z; bz = lo.w; }
        hits += __float_as_int(hi.x);
    }
    const float sgn = (hits & 1) ? -1.0f : 1.0f;
    out[i] = sqrtf(best) * sgn;
    out[N + 3*i + 0] = bx;
    out[N + 3*i + 1] = by;
    out[N + 3*i + 2] = bz;
}

// ---------- fallback: single pass (used only if workspace is too small) ----------
__global__ void sdf_single(const float* __restrict__ p, const float* __restrict__ v,
                           const int* __restrict__ f, float* __restrict__ out,
                           int N, int V, int F) {
    __shared__ __align__(16) float sface[TILE * 16];
    const int t = threadIdx.x;

    // per-block vertex mean (reuse sface as reduction scratch)
    {
        float ax = 0.f, ay = 0.f, az = 0.f;
        for (int j = t; j < V; j += BLK) { ax += v[3*j]; ay += v[3*j+1]; az += v[3*j+2]; }
        sface[t] = ax; sface[BLK + t] = ay; sface[2*BLK + t] = az;
        __syncthreads();
        for (int off = BLK / 2; off > 0; off >>= 1) {
            if (t < off) {
                sface[t]         += sface[t + off];
                sface[BLK + t]   += sface[BLK + t + off];
                sface[2*BLK + t] += sface[2*BLK + t + off];
            }
            __syncthreads();
        }
    }
    const float mnx = sface[0] / (float)V;
    const float mny = sface[BLK] / (float)V;
    const float mnz = sface[2*BLK] / (float)V;
    __syncthreads();

    const int i = blockIdx.x * BLK + t;
    const bool valid = (i < N);
    const float px = valid ? p[3*i]   : 0.f;
    const float py = valid ? p[3*i+1] : 0.f;
    const float pz = valid ? p[3*i+2] : 0.f;
    const float dvx = mnx - px, dvy = mny - py, dvz = mnz - pz;

    float best = __builtin_inff(), bcx = 0.f, bcy = 0.f, bcz = 0.f;
    int hits = 0;
    for (int faceBase = 0; faceBase < F; faceBase += TILE) {
        int cnt = F - faceBase; if (cnt > TILE) cnt = TILE;
        load_face_tile(v, f, faceBase, cnt, sface);
        process_tile(sface, cnt, px, py, pz, dvx, dvy, dvz, best, bcx, bcy, bcz, hits);
        __syncthreads();   // all readers done before next tile overwrites LDS
    }
    if (valid) {
        const float sgn = (hits & 1) ? -1.0f : 1.0f;
        out[i] = sqrtf(best) * sgn;
        out[N + 3*i + 0] = bcx;
        out[N + 3*i + 1] = bcy;
        out[N + 3*i + 2] = bcz;
    }
}

extern "C" void kernel_launch(void* const* d_in, const int* in_sizes, int n_in,
                              void* d_out, int out_size, void* d_ws, size_t ws_size,
                              hipStream_t stream) {
    const float* p = (const float*)d_in[0];
    const float* v = (const float*)d_in[1];
    const int*   f = (const int*)d_in[2];
    float* out = (float*)d_out;
    float* ws  = (float*)d_ws;

    const int N = in_sizes[0] / 3;   // 2048
    const int V = in_sizes[1] / 3;   // 2500
    const int F = in_sizes[2] / 3;   // 4096

    const int npb = (N + BLK - 1) / BLK;                 // 8 point blocks

    // Pick the finest face split whose partial records fit the workspace.
    // fbs=128 -> 8x32 = 256 workgroups (best WGP coverage), fbs=256 -> 128 WGs.
    int fbs = 0, nfb = 0;
    const int cand[2] = {128, 256};
    for (int k = 0; k < 2; ++k) {
        const int c  = cand[k];
        const int nb = (F + c - 1) / c;
        const size_t need = (16 + (size_t)N * nb * 8) * sizeof(float);
        if (ws_size >= need) { fbs = c; nfb = nb; break; }
    }

    if (fbs != 0) {
        sdf_vmean<<<1, BLK, 0, stream>>>(v, ws, V);
        dim3 grid(npb, nfb);
        sdf_pass1<<<grid, BLK, 0, stream>>>(p, v, f, ws, N, F, fbs, nfb);
        sdf_pass2<<<npb, BLK, 0, stream>>>(ws, out, N, nfb);
    } else {
        sdf_single<<<npb, BLK, 0, stream>>>(p, v, f, out, N, V, F);
    }
}